// GraphAttentionLayerForSingleNode_86199993631353
// MI455X (gfx1250) — compile-verified
//
#include <hip/hip_runtime.h>
#include <math.h>

typedef __attribute__((ext_vector_type(2))) float v2f;
typedef __attribute__((ext_vector_type(8))) float v8f;

#define B_TOT 16384
#define NN 64
#define DD 64
#define OO 64

// W pair-interleaved LDS index: rows d, d+1 interleaved per column so a
// B-fragment (K=d,d+1 at one column) is a single aligned 8-byte LDS load.
#define WPIDX(d, o) ((((d) >> 1) << 7) + ((o) << 1) + ((d) & 1))

// 8 waves / block, 1 batch element per wave.
__launch_bounds__(256)
__global__ void gat_wmma_f32_kernel(const float* __restrict__ h,
                                    const float* __restrict__ W,
                                    const float* __restrict__ a,
                                    float* __restrict__ out)
{
    __shared__ float Wl[DD * OO];      // 16 KB: W staged pair-interleaved
    __shared__ float c1l[DD];          // W @ a1
    __shared__ float c2l[DD];          // W @ a2
    __shared__ float fac[8 * NN];      // per-wave row scale factors (attn weights)

    const int tid  = threadIdx.x;
    const int lane = tid & 31;         // wave32 (gfx1250)
    const int wv   = tid >> 5;

    // ---- stage W into LDS, pair-interleaving K rows ----
    for (int idx = tid; idx < DD * OO; idx += 256) {
        const int d = idx >> 6;
        const int o = idx & 63;
        Wl[WPIDX(d, o)] = W[idx];      // coalesced global b32 reads
    }
    __syncthreads();

    // ---- c1 = W@a1 , c2 = W@a2 (once per block) ----
    if (tid < 128) {
        const int d = tid & 63;
        const float* av = a + ((tid < 64) ? 0 : OO);
        float s = 0.f;
        #pragma unroll 8
        for (int o = 0; o < OO; ++o) s = fmaf(Wl[WPIDX(d, o)], av[o], s);
        if (tid < 64) c1l[d] = s; else c2l[d] = s;
    }
    __syncthreads();

    const int b = blockIdx.x * 8 + wv;
    const float* __restrict__ hb = h + (size_t)b * (NN * DD);
    float* __restrict__ ob = out + (size_t)b * (NN * OO);

    // prefetch this batch element's h tile (16 KB, 512B per lane)
    __builtin_prefetch(hb + lane * 128, 0, 0);

    // ---- scores: s[n] = h[n]·c2 + h[0]·c1 ; softmax over n=1..63 ----
    float s0 = 0.f, s1 = 0.f, r = 0.f;
    #pragma unroll 4
    for (int d = 0; d < DD; ++d) {
        const float cc2 = c2l[d];
        s0 = fmaf(hb[lane * DD + d],        cc2, s0);   // row = lane
        s1 = fmaf(hb[(lane + 32) * DD + d], cc2, s1);   // row = lane + 32
        r  = fmaf(hb[d], c1l[d], r);                    // robot row (uniform)
    }
    const float t0 = s0 + r;
    const float t1 = s1 + r;

    float mx = (lane == 0) ? t1 : fmaxf(t0, t1);        // row 0 excluded
    #pragma unroll
    for (int off = 16; off > 0; off >>= 1)
        mx = fmaxf(mx, __shfl_xor(mx, off, 32));

    const float e0 = (lane == 0) ? 0.f : expf(t0 - mx);
    const float e1 = expf(t1 - mx);
    float sm = e0 + e1;
    #pragma unroll
    for (int off = 16; off > 0; off >>= 1)
        sm += __shfl_xor(sm, off, 32);
    const float inv = 1.f / sm;

    fac[wv * NN + lane]      = (lane == 0) ? 1.f : e0 * inv;  // row 0 = robot, unscaled
    fac[wv * NN + 32 + lane] = e1 * inv;
    __syncthreads();

    // ---- GEMM: Wh = h_b @ W via V_WMMA_F32_16X16X4_F32, fused scale + ELU ----
    // A frag (16x4 f32): lane L holds row M=L%16, K = 2*(L/16)+{0,1}  -> aligned float2 (global b64)
    // B frag (4x16 f32): lane L holds col N=L%16, K = 2*(L/16)+{0,1}  -> one aligned LDS b64
    // C/D  (16x16 f32): VGPR r, lane L -> M = r + 8*(L/16), N = L%16
    const int mrow  = lane & 15;
    const int khalf = lane >> 4;

    for (int mt = 0; mt < 4; ++mt) {
        v8f acc[4];
        #pragma unroll
        for (int nt = 0; nt < 4; ++nt) acc[nt] = (v8f){0.f,0.f,0.f,0.f,0.f,0.f,0.f,0.f};

        const float* __restrict__ arow = hb + (mt * 16 + mrow) * DD + 2 * khalf;

        #pragma unroll
        for (int k = 0; k < 16; ++k) {
            const v2f af = *(const v2f*)(arow + k * 4);
            // pair-row index for K = k*4 + 2*khalf (always even): (K>>1) = k*2 + khalf
            const float* __restrict__ wk = Wl + ((k * 2 + khalf) << 7) + (mrow << 1);
            #pragma unroll
            for (int nt = 0; nt < 4; ++nt) {
                const v2f bf = *(const v2f*)(wk + (nt << 5));   // single ds_load_b64
                acc[nt] = __builtin_amdgcn_wmma_f32_16x16x4_f32(
                    /*neg_a=*/false, af, /*neg_b=*/false, bf,
                    /*c_mod=*/(short)0, acc[nt],
                    /*reuse_a=*/false, /*reuse_b=*/false);
            }
        }

        // epilogue: out[row] = elu( fac[row] * Wh[row] )
        #pragma unroll
        for (int nt = 0; nt < 4; ++nt) {
            #pragma unroll
            for (int rr = 0; rr < 8; ++rr) {
                const int row = mt * 16 + rr + 8 * khalf;
                float v = acc[nt][rr] * fac[wv * NN + row];
                v = (v > 0.f) ? v : expm1f(v);
                ob[row * OO + nt * 16 + mrow] = v;
            }
        }
    }
}

extern "C" void kernel_launch(void* const* d_in, const int* in_sizes, int n_in,
                              void* d_out, int out_size, void* d_ws, size_t ws_size,
                              hipStream_t stream) {
    (void)in_sizes; (void)n_in; (void)out_size; (void)d_ws; (void)ws_size;
    const float* h = (const float*)d_in[0];   // (16384, 64, 64) f32
    const float* W = (const float*)d_in[1];   // (64, 64) f32
    const float* a = (const float*)d_in[2];   // (128, 1) f32
    float* out = (float*)d_out;               // (16384, 64, 64) f32

    dim3 grid(B_TOT / 8);   // 2048 blocks, 8 waves each -> one batch element per wave
    gat_wmma_f32_kernel<<<grid, 256, 0, stream>>>(h, W, a, out);
}